// NeuronPool_26843545600728
// MI455X (gfx1250) — compile-verified
//
#include <hip/hip_runtime.h>
#include <math.h>
#include <stdint.h>

// ---------------------------------------------------------------------------
// NeuronPool on MI455X (gfx1250): 64 neurons x (2304->512->512->256) MLP,
// batch 32.  Memory-bound (~403 MB fp32 weights / call, AI ~16 F/B).
// Strategy: stream W tiles with the Tensor Data Mover (tensor_load_to_lds,
// double-buffered, TENSORcnt-tracked) and feed V_WMMA_F32_16X16X4_F32.
// ---------------------------------------------------------------------------

typedef __attribute__((ext_vector_type(2))) float v2f;
typedef __attribute__((ext_vector_type(8))) float v8f;
typedef __attribute__((ext_vector_type(4))) unsigned int v4u;
typedef __attribute__((ext_vector_type(8))) unsigned int v8u;

#define MODEL_DIM 256
#define HIST      8
#define HID       512
#define N_NEURONS 64
#define IN_DIM    (MODEL_DIM * (1 + HIST))   // 2304
#define BATCH     32

#define BM 32        // batch tile (full batch)
#define BN 64        // output-column tile per workgroup
#define BK 64        // K chunk staged in LDS
#define XS_LD 68     // padded LDS stride for X tile (conflict-free column reads)
#define WS_LD 72     // padded LDS stride for W tile (TDM pad: 8 dwords / 64 dwords)

// ---------------------------------------------------------------------------
// Kernel A: build x = [emb @ Wp + bp  |  broadcast(hist)]   (32 x 2304)
// ---------------------------------------------------------------------------
__global__ __launch_bounds__(256) void np_build_x(
    const float* __restrict__ emb,   // (32,256)
    const float* __restrict__ Wp,    // (256,256)
    const float* __restrict__ bp,    // (256)
    const float* __restrict__ hist,  // (8,256) flat 2048
    float* __restrict__ x)           // (32,2304)
{
    int idx = blockIdx.x * 256 + threadIdx.x;      // 32*2304 = 73728
    int b = idx / IN_DIM;
    int c = idx - b * IN_DIM;
    if (c < MODEL_DIM) {
        float s = bp[c];
        const float* er = emb + b * MODEL_DIM;
        #pragma unroll 4
        for (int k = 0; k < MODEL_DIM; ++k)
            s = fmaf(er[k], Wp[k * MODEL_DIM + c], s);
        x[idx] = s;
    } else {
        x[idx] = hist[c - MODEL_DIM];
    }
}

// ---------------------------------------------------------------------------
// Generic per-neuron WMMA GEMM with TDM-streamed, double-buffered W tiles:
//   Y[n] (32 x N) = act( X[n](32 x K) @ W[n](K x N) + b[n] )
//   grid.x = N_NEURONS * (N / BN), block = 256 (8 wave32s -> 2x4 16x16 tiles)
//   xStride == 0 -> X shared across neurons (GEMM1); else X + n*xStride.
// ---------------------------------------------------------------------------
__global__ __launch_bounds__(256) void np_gemm(
    const float* __restrict__ X,
    const float* __restrict__ W,     // (N_NEURONS, K, N) row-major
    const float* __restrict__ bias,  // (N_NEURONS, N)
    float* __restrict__ Y,           // (N_NEURONS, 32, N)
    int K, int N, int nblk, int applyGelu, long long xStride)
{
    const int n  = blockIdx.x / nblk;           // neuron
    const int nb = blockIdx.x - n * nblk;       // column block

    const float* Xn = X + (long long)n * xStride;                    // 32 x K, ld = K
    const float* Wn = W + (long long)n * (long long)K * (long long)N;

    __shared__ float Xs[BM][XS_LD];
    __shared__ float Ws[2][BK][WS_LD];          // TDM double buffer (padded rows)

    const int t    = threadIdx.x;
    const int lane = t & 31;
    const int wave = t >> 5;
    const int mi   = wave & 1;        // 0..1  -> M tile
    const int ni   = wave >> 1;       // 0..3  -> N tile
    const int ml   = lane & 15;
    const int half = lane >> 4;       // 0/1
    const int kp   = half << 1;       // K sub-offset 0 or 2 (ISA A/B lane layout)
    const int m    = mi * 16 + ml;    // batch row for A fragment
    const int nn   = ni * 16 + ml;    // local column for B fragment

    // TDM descriptor group1: data_size=4B (2<<16), pad_enable (1<<20),
    // pad_interval=5 -> 64 dwords (22:24), pad_amount=7 -> 8 dwords (25:31),
    // tensor_dim0=N, tensor_dim1=K, tile 64x64, dim0_stride=N.
    const unsigned uN = (unsigned)N, uK = (unsigned)K;
    const v8u g1 = { (2u << 16) | (1u << 20) | (5u << 22) | (7u << 25),
                     (uN & 0xFFFFu) << 16,
                     (uN >> 16) | ((uK & 0xFFFFu) << 16),
                     (uK >> 16) | ((unsigned)BN << 16),
                     (unsigned)BK,
                     uN, 0u, 0u };
    const v4u gz = { 0u, 0u, 0u, 0u };

    // issue one TDM tile load: rows [kload, kload+BK) x cols [nb*BN, nb*BN+BN)
    auto tdm_issue = [&](int kload, int p) {
        uint64_t ga = (uint64_t)(uintptr_t)(Wn + (size_t)kload * N + (size_t)nb * BN);
        uint32_t lds = (uint32_t)(uintptr_t)&Ws[p][0][0];
        v4u g0 = { 1u,                                   // count=1 valid descriptor
                   lds,                                  // lds_addr
                   (uint32_t)ga,                         // global_addr[31:0]
                   (uint32_t)(ga >> 32) | 0x80000000u }; // addr[56:32] | type=2
        asm volatile("tensor_load_to_lds %0, %1, %2, %3"
                     :
                     : "s"(g0), "s"(g1), "s"(gz), "s"(gz)
                     : "memory");
    };

    v8f acc = {};
    const int nch = K / BK;

    if (wave == 0) tdm_issue(0, 0);             // prologue: chunk 0 -> buf 0

    for (int ci = 0; ci < nch; ++ci) {
        const int k0  = ci * BK;
        const int cur = ci & 1;

        // issue next W chunk into the other buffer (readers of it finished at
        // the trailing barrier of the previous iteration)
        if (wave == 0 && ci + 1 < nch) tdm_issue(k0 + BK, cur ^ 1);

        // stage X tile: 32 x 64 floats (512 float4, 2 per thread)
        #pragma unroll
        for (int i = t; i < BM * (BK / 4); i += 256) {
            int r  = i >> 4;
            int c4 = (i & 15) << 2;
            float4 v = *(const float4*)(Xn + (long long)r * K + k0 + c4);
            *(float4*)&Xs[r][c4] = v;
        }
        if (ci + 1 < nch)
            __builtin_prefetch(Xn + (long long)(t & 31) * K + k0 + BK, 0, 1);

        // wave0: current chunk's DMA must be complete (next may stay in flight)
        if (wave == 0) {
            if (ci + 1 < nch) __builtin_amdgcn_s_wait_tensorcnt(1);
            else              __builtin_amdgcn_s_wait_tensorcnt(0);
        }
        __syncthreads();   // publish Xs + Ws[cur] to all waves

        // ---- 16 x V_WMMA_F32_16X16X4_F32 per chunk ----
        #pragma unroll
        for (int kk = 0; kk < BK; kk += 4) {
            v2f a = { Xs[m][kk + kp], Xs[m][kk + kp + 1] };
            v2f b = { Ws[cur][kk + kp][nn], Ws[cur][kk + kp + 1][nn] };
            acc = __builtin_amdgcn_wmma_f32_16x16x4_f32(
                false, a, false, b, (short)0, acc, false, false);
        }
        __syncthreads();   // all reads done before buffers are overwritten
    }

    // ---- epilogue: +bias, optional exact GELU, scatter per ISA C layout ----
    const int cn = nb * BN + nn;                       // global column
    const float bv = bias[(long long)n * N + cn];
    float* Yn = Y + ((long long)n * BM) * N;
    #pragma unroll
    for (int r = 0; r < 8; ++r) {
        int cm = mi * 16 + r + 8 * half;               // batch row
        float v = acc[r] + bv;
        if (applyGelu)
            v = 0.5f * v * (1.0f + erff(v * 0.70710678118654752f));
        Yn[(long long)cm * N + cn] = v;
    }
}

// ---------------------------------------------------------------------------
// Kernel E: per-(b,n) LayerNorm over D=256 + oscillator modulation.
//   grid = BATCH * N_NEURONS blocks, 256 threads each.
// ---------------------------------------------------------------------------
__global__ __launch_bounds__(256) void np_ln_mod(
    const float* __restrict__ yraw,   // (N_NEURONS, 32, 256)
    const float* __restrict__ gamma,  // (64,256)
    const float* __restrict__ beta,   // (64,256)
    const int*   __restrict__ tick,
    float* __restrict__ out)          // (32,64,256)
{
    const int blk = blockIdx.x;           // 0..2047
    const int b = blk >> 6;
    const int n = blk & 63;
    const int t = threadIdx.x;
    const int lane = t & 31;
    const int wave = t >> 5;

    const float v = yraw[(((long long)n * BATCH) + b) * MODEL_DIM + t];

    // wave-level reduction of sum / sumsq, then across 8 waves via LDS
    float s = v, ss = v * v;
    #pragma unroll
    for (int off = 16; off > 0; off >>= 1) {
        s  += __shfl_xor(s,  off, 32);
        ss += __shfl_xor(ss, off, 32);
    }
    __shared__ float sbuf[8], ssbuf[8];
    if (lane == 0) { sbuf[wave] = s; ssbuf[wave] = ss; }
    __syncthreads();
    float tot = 0.f, tot2 = 0.f;
    #pragma unroll
    for (int i = 0; i < 8; ++i) { tot += sbuf[i]; tot2 += ssbuf[i]; }

    const float mu  = tot * (1.0f / MODEL_DIM);
    const float var = tot2 * (1.0f / MODEL_DIM) - mu * mu;
    const float inv = rsqrtf(var + 1e-5f);

    // oscillator: freq = 0.5 * 80^(n/63), phase = (2.3571*n) mod 2pi
    const float fi    = (float)n;
    const float freq  = 0.5f * __expf(fi * (1.0f / 63.0f) * 4.38202663467388f); // ln(80)
    const float phase = fmodf(fi * 2.3571f, 6.283185307179586f);
    const float tt    = (float)tick[0] * 0.1f;
    const float mod   = 1.0f + 0.5f * __sinf(6.283185307179586f * freq * tt + phase);

    const float g = gamma[n * MODEL_DIM + t];
    const float be = beta[n * MODEL_DIM + t];
    out[(((long long)b * N_NEURONS) + n) * MODEL_DIM + t] = ((v - mu) * inv * g + be) * mod;
}

// ---------------------------------------------------------------------------
extern "C" void kernel_launch(void* const* d_in, const int* in_sizes, int n_in,
                              void* d_out, int out_size, void* d_ws, size_t ws_size,
                              hipStream_t stream) {
    const float* emb   = (const float*)d_in[0];
    const float* hist  = (const float*)d_in[1];
    const float* Wp    = (const float*)d_in[2];
    const float* bp    = (const float*)d_in[3];
    const float* W1    = (const float*)d_in[4];
    const float* b1    = (const float*)d_in[5];
    const float* W2    = (const float*)d_in[6];
    const float* b2    = (const float*)d_in[7];
    const float* W3    = (const float*)d_in[8];
    const float* b3    = (const float*)d_in[9];
    const float* gamma = (const float*)d_in[10];
    const float* beta  = (const float*)d_in[11];
    const int*   tick  = (const int*)d_in[12];
    float* out = (float*)d_out;

    // workspace layout (floats): x | h1 | h2 | yraw   (~10.8 MB total)
    float* xbuf = (float*)d_ws;                                   // 32*2304
    float* h1   = xbuf + (long long)BATCH * IN_DIM;               // 64*32*512
    float* h2   = h1   + (long long)N_NEURONS * BATCH * HID;      // 64*32*512
    float* yraw = h2   + (long long)N_NEURONS * BATCH * HID;      // 64*32*256

    // 1) x = [emb@Wp + bp | hist]
    np_build_x<<<(BATCH * IN_DIM) / 256, 256, 0, stream>>>(emb, Wp, bp, hist, xbuf);

    // 2) h1 = gelu(x @ W1 + b1)      K=2304, N=512, x shared across neurons
    np_gemm<<<N_NEURONS * (HID / BN), 256, 0, stream>>>(
        xbuf, W1, b1, h1, IN_DIM, HID, HID / BN, 1, 0LL);

    // 3) h2 = gelu(h1 @ W2 + b2)     K=512, N=512, per-neuron X
    np_gemm<<<N_NEURONS * (HID / BN), 256, 0, stream>>>(
        h1, W2, b2, h2, HID, HID, HID / BN, 1, (long long)BATCH * HID);

    // 4) y = h2 @ W3 + b3            K=512, N=256
    np_gemm<<<N_NEURONS * (MODEL_DIM / BN), 256, 0, stream>>>(
        h2, W3, b3, yraw, HID, MODEL_DIM, MODEL_DIM / BN, 0, (long long)BATCH * HID);

    // 5) LayerNorm + oscillator modulation
    np_ln_mod<<<BATCH * N_NEURONS, 256, 0, stream>>>(yraw, gamma, beta, tick, out);
}